// ModularQFM_73804718014526
// MI455X (gfx1250) — compile-verified
//
#include <hip/hip_runtime.h>

// ModularQFM on MI455X (gfx1250):
//  - real-valued 16-qubit statevector (all gates real => complex part provably 0)
//  - one workgroup per batch element, full 256KB state resident in LDS (320KB/WGP)
//  - per layer: 4 fused 16x16 quad-gate matrices (RY x4 + intra-quad CNOTs),
//    applied as tiled 16x16x16 real matmuls via chained V_WMMA_F32_16X16X4_F32
//  - 3 cross-quad CNOTs per ladder as one in-place LDS swap pass
//  - expectations as 4x4 symmetric quadratic forms + WG reduction, tiny head
// Zero HBM traffic for the state => compute/LDS bound.
// Round-2 refinements: scalar (SGPR) tile-loop control with compile-time trip
// count so EXEC is provably all-ones around every WMMA, and ds_load_b64 for
// the contiguous quad-3 A-operand pairs.

typedef __attribute__((ext_vector_type(2))) float v2f;
typedef __attribute__((ext_vector_type(8))) float v8f;

#define NQ      16
#define NSTATE  65536     // 2^16 real amplitudes
#define WG      512       // 16 waves (wave32)
#define NWAVES  (WG / 32)
#define TILES   256       // 16x16 tiles covering the state per quad pass
#define TPW     (TILES / NWAVES)
#define NOBS    15

__device__ __forceinline__ v8f wmma_f32_16x16x4(v2f a, v2f b, v8f c) {
  // (neg_a, A, neg_b, B, c_mod, C, reuse_a, reuse_b)
  return __builtin_amdgcn_wmma_f32_16x16x4_f32(false, a, false, b, (short)0, c,
                                               false, false);
}

__global__ __launch_bounds__(WG, 1)
void qfm_kernel(const float* __restrict__ x,    // (256,64) encoding angles
                const float* __restrict__ A,    // (15,6) Re off-diag of H
                const float* __restrict__ D,    // (15,4) diag seeds
                const float* __restrict__ vp,   // (4,16) shared VQC angles
                const float* __restrict__ hw,   // (64,15) head weight
                const float* __restrict__ hb,   // (64,) head bias
                float* __restrict__ out) {      // (256,64)
  extern __shared__ float smem[];
  float* S   = smem;              // [65536] state
  float* U   = smem + NSTATE;     // [4*16*16] fused quad gate matrices
  float* csn = U + 1024;          // [32] cos/sin of theta/2 for current layer
  float* ew  = csn + 32;          // [15] expectation values
  float* red = U;                 // reduction scratch (reuses U after gates)

  const int tid  = threadIdx.x;
  const int lane = tid & 31;
  // SGPR wave id -> scalar loop control, EXEC untouched around WMMA
  const int wave = __builtin_amdgcn_readfirstlane(tid >> 5);
  const int b    = blockIdx.x;

  const int lane15 = lane & 15;
  const int koff   = (lane < 16) ? 0 : 2;  // A/B operand K split across half-waves
  const int hi8    = (lane < 16) ? 0 : 8;  // C/D operand M split

  // ---- init |0...0> ----
  for (int i = tid; i < NSTATE; i += WG) S[i] = 0.0f;
  if (tid == 0) S[0] = 1.0f;
  __syncthreads();

  // ---- 8 fused gate layers: 4 encoding (pass 0..3) + 4 VQC (pass 4..7) ----
  for (int pass = 0; pass < 8; ++pass) {
    const bool ladder = (pass != 3);  // last encoding layer has no CNOT ladder

    // cos/sin(theta/2) for this layer's 16 angles
    if (tid < NQ) {
      float th = (pass < 4) ? x[b * 64 + pass * 16 + tid]
                            : vp[(pass - 4) * 16 + tid];
      float sv, cv;
      __sincosf(th * 0.5f, &sv, &cv);
      csn[tid] = cv;
      csn[16 + tid] = sv;
    }
    __syncthreads();

    // Build fused 16x16 quad matrices: U = P_odd * P_even * (Ry x Ry x Ry x Ry)
    // (even CNOTs and the odd CNOT(4g+1,4g+2) are intra-quad permutations)
    for (int e = tid; e < 1024; e += WG) {
      int g  = e >> 8;         // quad
      int rp = (e >> 4) & 15;  // output row r'
      int cc = e & 15;         // column
      int r = rp;
      if (ladder) {
        if (r & 4) r ^= 2;     // sigma_odd : CNOT(q1,q2)
        if (r & 8) r ^= 4;     // sigma_even: CNOT(q0,q1)
        if (r & 2) r ^= 1;     //             CNOT(q2,q3)
      }
      float val = 1.0f;
      #pragma unroll
      for (int j = 0; j < 4; ++j) {            // qubit 4g+j, MSB first
        int rb = (r  >> (3 - j)) & 1;
        int cb = (cc >> (3 - j)) & 1;
        float c_ = csn[4 * g + j];
        float s_ = csn[16 + 4 * g + j];
        // Ry = [[c,-s],[s,c]]
        val *= (rb == cb) ? c_ : ((rb == 1) ? s_ : -s_);
      }
      U[e] = val;
    }
    __syncthreads();

    // Apply quad matrices: state as (pre, 16, post); 256 disjoint 16x16 tiles
    for (int g = 0; g < 4; ++g) {
      const float* Ug = U + (g << 8);

      // Preload the constant U operand slices (reused for all tiles this wave).
      // quads 0-2: A-operand A[m][k]=U[m][k]; quad 3: B-operand Ut[k][n]=U[n][k]
      // -> identical per-lane gather either way (8B-aligned pair loads).
      v2f uop[4];
      #pragma unroll
      for (int kc = 0; kc < 4; ++kc) {
        int k = 4 * kc + koff;
        uop[kc] = *(const v2f*)(Ug + lane15 * 16 + k);
      }

      if (g < 3) {
        const int postshift = 12 - 4 * g;       // post = 4096, 256, 16
        const int post = 1 << postshift;
        const int ntc = post >> 4;              // 16-col tiles per pre-slab
        #pragma unroll 2
        for (int tt = 0; tt < TPW; ++tt) {      // scalar trip count
          int t = tt * NWAVES + wave;           // SGPR tile id
          int pre = t / ntc;
          int colblk = t - pre * ntc;
          int base = (pre << (4 + postshift)) + (colblk << 4);
          v8f acc = {0.f, 0.f, 0.f, 0.f, 0.f, 0.f, 0.f, 0.f};
          #pragma unroll
          for (int kc = 0; kc < 4; ++kc) {      // D += U[:,4kc:4kc+4] * S-rows
            int k0 = 4 * kc + koff;
            v2f bop;
            bop[0] = S[base + k0 * post + lane15];
            bop[1] = S[base + (k0 + 1) * post + lane15];
            acc = wmma_f32_16x16x4(uop[kc], bop, acc);
          }
          #pragma unroll
          for (int vv = 0; vv < 8; ++vv)        // in-place: tile is closed
            S[base + (vv + hi8) * post + lane15] = acc[vv];
        }
      } else {
        // quad 3: post==1; state as (4096,16) rows, compute S' = S * U^T
        #pragma unroll 2
        for (int tt = 0; tt < TPW; ++tt) {
          int t = tt * NWAVES + wave;
          int base = t << 8;                    // 16 rows x 16 contiguous amps
          v8f acc = {0.f, 0.f, 0.f, 0.f, 0.f, 0.f, 0.f, 0.f};
          #pragma unroll
          for (int kc = 0; kc < 4; ++kc) {
            int k0 = 4 * kc + koff;
            // contiguous, 8B-aligned pair -> single ds_load_b64
            v2f aop = *(const v2f*)(S + base + lane15 * 16 + k0);
            acc = wmma_f32_16x16x4(aop, uop[kc], acc);
          }
          #pragma unroll
          for (int vv = 0; vv < 8; ++vv)
            S[base + (vv + hi8) * 16 + lane15] = acc[vv];
        }
      }
      __syncthreads();
    }

    // Cross-quad CNOTs (3,4),(7,8),(11,12) -> bit pairs (12,11),(8,7),(4,3).
    // Disjoint commuting involution: one in-place swap pass.
    if (ladder) {
      for (int i = tid; i < NSTATE; i += WG) {
        int j = i;
        if (i & (1 << 12)) j ^= (1 << 11);
        if (i & (1 << 8))  j ^= (1 << 7);
        if (i & (1 << 4))  j ^= (1 << 3);
        if (j > i) { float tv = S[i]; S[i] = S[j]; S[j] = tv; }
      }
      __syncthreads();
    }
  }

  // ---- expectations: e_w = sum over groups  p^T Re(H_w) p,  p in R^4 ----
  for (int w = 0; w < NOBS; ++w) {
    const int pb = 14 - w;            // bit of qubit w+1 (qubit w = pb+1)
    const int maskA = 1 << (pb + 1);  // qubit w  (MSB of the 4-index)
    const int maskB = 1 << pb;        // qubit w+1
    // Re(H): diag = 2*D[1..3] (H[3][3]=0); off-diag = A in tril order
    const float d1 = 2.0f * D[w * 4 + 1];
    const float d2 = 2.0f * D[w * 4 + 2];
    const float d3 = 2.0f * D[w * 4 + 3];
    const float a0 = A[w * 6 + 0], a1 = A[w * 6 + 1], a2 = A[w * 6 + 2];
    const float a3 = A[w * 6 + 3], a4 = A[w * 6 + 4], a5 = A[w * 6 + 5];
    float part = 0.0f;
    for (int m = tid; m < 16384; m += WG) {
      int basei = ((m >> pb) << (pb + 2)) | (m & (maskB - 1));
      float p00 = S[basei];
      float p01 = S[basei | maskB];
      float p10 = S[basei | maskA];
      float p11 = S[basei | maskA | maskB];
      part += d1 * p00 * p00 + d2 * p01 * p01 + d3 * p10 * p10 +
              2.0f * (a0 * p01 * p00 + a1 * p10 * p00 + a2 * p10 * p01 +
                      a3 * p11 * p00 + a4 * p11 * p01 + a5 * p11 * p10);
    }
    red[tid] = part;
    __syncthreads();
    for (int s = WG / 2; s > 0; s >>= 1) {
      if (tid < s) red[tid] += red[tid + s];
      __syncthreads();
    }
    if (tid == 0) ew[w] = red[0];
    __syncthreads();
  }

  // ---- head: out[b,o] = hb[o] + sum_w ew[w] * hw[o,w] ----
  if (tid < 64) {
    float acc = hb[tid];
    #pragma unroll
    for (int w = 0; w < NOBS; ++w) acc += ew[w] * hw[tid * NOBS + w];
    out[b * 64 + tid] = acc;
  }
}

extern "C" void kernel_launch(void* const* d_in, const int* in_sizes, int n_in,
                              void* d_out, int out_size, void* d_ws,
                              size_t ws_size, hipStream_t stream) {
  (void)in_sizes; (void)n_in; (void)out_size; (void)d_ws; (void)ws_size;
  const float* x  = (const float*)d_in[0];
  const float* A  = (const float*)d_in[1];
  // d_in[2] = B (imag off-diagonals): unused — state is real, Im(H) cancels.
  const float* D  = (const float*)d_in[3];
  const float* vp = (const float*)d_in[4];
  const float* hw = (const float*)d_in[5];
  const float* hb = (const float*)d_in[6];
  float* out = (float*)d_out;

  const size_t shmem =
      (size_t)(NSTATE + 1024 + 32 + 16) * sizeof(float);  // ~261 KB < 320 KB/WGP
  qfm_kernel<<<256, WG, shmem, stream>>>(x, A, D, vp, hw, hb, out);
}